// RoiPoolingConv_88862873354943
// MI455X (gfx1250) — compile-verified
//
#include <hip/hip_runtime.h>

typedef float f32x4 __attribute__((ext_vector_type(4)));

#define IMG_H 200
#define IMG_W 272
#define IMG_C 1024
#define POOL  7
#define TPB   256

// One workgroup per (roi, py). Each thread owns 4 channels (one b128 lane).
// Corner gathers go through the CDNA5 async-to-LDS path, double-buffered
// across the 7 output pixels; ASYNCcnt tracks completion.
__global__ __launch_bounds__(TPB) void roi_pool_async_kernel(
    const float* __restrict__ img, const int* __restrict__ rois,
    float* __restrict__ out)
{
    __shared__ f32x4 stage[2][4][TPB];   // 2 bufs x 4 corners x 256 lanes = 32 KB

    const int t   = threadIdx.x;
    const int bid = blockIdx.x;          // r*POOL + py
    const int r   = bid / POOL;
    const int py  = bid - r * POOL;

    const int rx = rois[r * 4 + 0];
    const int ry = rois[r * 4 + 1];
    const int rw = rois[r * 4 + 2];
    const int rh = rois[r * 4 + 3];

    const float hf = (float)rh;
    const float wf = (float)rw;

    // Vertical source row (uniform across the block) — mirrors the reference:
    // ys = y + clip((i+0.5)*h/7 - 0.5, 0, h-1); y0 = floor(ys); y1 = min(y0+1, y+h-1)
    float cy = ((float)py + 0.5f) * hf * (1.0f / POOL) - 0.5f;
    cy = fminf(fmaxf(cy, 0.0f), hf - 1.0f);
    const float ysf = (float)ry + cy;
    const int   y0  = (int)floorf(ysf);
    const float wy  = ysf - (float)y0;
    const int   y1  = min(y0 + 1, ry + rh - 1);

    int   x0a[POOL], x1a[POOL];
    float wxa[POOL];
#pragma unroll
    for (int px = 0; px < POOL; ++px) {
        float cx = ((float)px + 0.5f) * wf * (1.0f / POOL) - 0.5f;
        cx = fminf(fmaxf(cx, 0.0f), wf - 1.0f);
        float xsf = (float)rx + cx;
        int   x0  = (int)floorf(xsf);
        wxa[px] = xsf - (float)x0;
        x0a[px] = x0;
        x1a[px] = min(x0 + 1, rx + rw - 1);
    }

    const int cb   = t * 16;                                    // channel byte offset
    const int row0 = y0 * (IMG_W * (int)sizeof(float) * IMG_C); // byte offset of row y0
    const int row1 = y1 * (IMG_W * (int)sizeof(float) * IMG_C);
    const unsigned long long base = (unsigned long long)img;

    auto issue = [&](int px, int buf) {
        const int c0  = x0a[px] * ((int)sizeof(float) * IMG_C);
        const int c1  = x1a[px] * ((int)sizeof(float) * IMG_C);
        const int o00 = row0 + c0 + cb;
        const int o01 = row0 + c1 + cb;
        const int o10 = row1 + c0 + cb;
        const int o11 = row1 + c1 + cb;
        unsigned l00 = (unsigned)(size_t)&stage[buf][0][t];
        unsigned l01 = (unsigned)(size_t)&stage[buf][1][t];
        unsigned l10 = (unsigned)(size_t)&stage[buf][2][t];
        unsigned l11 = (unsigned)(size_t)&stage[buf][3][t];
        // GVS mode: address is the 64-bit SGPR base plus the signed 32-bit VGPR offset plus the immediate offset; LDS dest addr per lane.
        asm volatile("global_load_async_to_lds_b128 %0, %1, %2"
                     :: "v"(l00), "v"(o00), "s"(base) : "memory");
        asm volatile("global_load_async_to_lds_b128 %0, %1, %2"
                     :: "v"(l01), "v"(o01), "s"(base) : "memory");
        asm volatile("global_load_async_to_lds_b128 %0, %1, %2"
                     :: "v"(l10), "v"(o10), "s"(base) : "memory");
        asm volatile("global_load_async_to_lds_b128 %0, %1, %2"
                     :: "v"(l11), "v"(o11), "s"(base) : "memory");
    };

    issue(0, 0);

    float* outrow = out + ((size_t)bid * POOL) * IMG_C;

#pragma unroll
    for (int p = 0; p < POOL; ++p) {
        if (p + 1 < POOL) {
            issue(p + 1, (p + 1) & 1);
            // 8 async loads in flight; completion is in-order, so <=4 means
            // pixel p's 4 corner tiles have landed in LDS.
            asm volatile("s_wait_asynccnt 0x4" ::: "memory");
        } else {
            asm volatile("s_wait_asynccnt 0x0" ::: "memory");
        }
        const int buf = p & 1;
        f32x4 c00 = stage[buf][0][t];
        f32x4 c01 = stage[buf][1][t];
        f32x4 c10 = stage[buf][2][t];
        f32x4 c11 = stage[buf][3][t];

        const float wx  = wxa[p];
        const float iwy = 1.0f - wy;
        const float iwx = 1.0f - wx;
        // vertical lerp first, then horizontal (matches reference order)
        f32x4 v0 = c00 * iwy + c10 * wy;
        f32x4 v1 = c01 * iwy + c11 * wy;
        f32x4 o  = v0 * iwx + v1 * wx;

        // Output is streamed once: non-temporal store keeps the 223 MB image
        // resident in the 192 MB L2 for inter-ROI reuse.
        __builtin_nontemporal_store(o, (f32x4*)(outrow + p * IMG_C) + t);
    }
}

extern "C" void kernel_launch(void* const* d_in, const int* in_sizes, int n_in,
                              void* d_out, int out_size, void* d_ws, size_t ws_size,
                              hipStream_t stream) {
    (void)n_in; (void)out_size; (void)d_ws; (void)ws_size;
    const float* img  = (const float*)d_in[0];
    const int*   rois = (const int*)d_in[1];
    float*       out  = (float*)d_out;

    const int R = in_sizes[1] / 4;       // rois is (1, R, 4)
    dim3 grid(R * POOL);                 // one block per (roi, output-row)
    roi_pool_async_kernel<<<grid, TPB, 0, stream>>>(img, rois, out);
}